// GCN_Transformer_66271345377743
// MI455X (gfx1250) — compile-verified
//
#include <hip/hip_runtime.h>
#include <math.h>

// ---------------- problem constants ----------------
#define NODES   81
#define IN_DIM  162
#define DMODEL  256
#define NH      16
#define HD      16
#define DFF     2048
#define TT      1024
#define BB      8
#define MROWS   (TT*BB)   // 8192

typedef __attribute__((ext_vector_type(2))) float v2f;
typedef __attribute__((ext_vector_type(8))) float v8f;

// Native f32 WMMA: D(16x16) = A(16x4) * B(4x16) + C, all f32.
// 8-arg pattern: (neg_a, A, neg_b, B, c_mod, C, reuse_a, reuse_b)
__device__ __forceinline__ v8f wmma_f32(v2f a, v2f b, v8f c) {
  return __builtin_amdgcn_wmma_f32_16x16x4_f32(false, a, false, b, (short)0, c,
                                               false, false);
}

// ---------------- A_hat = D^-1/2 (A + I) D^-1/2  (81x81) ----------------
__global__ void __launch_bounds__(256) build_ahat(const int* __restrict__ ei,
                                                  int E, float* __restrict__ Ahat) {
  __shared__ float A[NODES * NODES];
  __shared__ float dinv[NODES];
  int tid = threadIdx.x;
  for (int i = tid; i < NODES * NODES; i += 256) A[i] = 0.f;
  __syncthreads();
  for (int e = tid; e < E; e += 256) {
    int r = ei[e];        // edge_index[0][e]
    int c = ei[E + e];    // edge_index[1][e]
    atomicAdd(&A[c * NODES + r], 1.0f);   // A[col,row] += 1
  }
  __syncthreads();
  for (int i = tid; i < NODES; i += 256) A[i * NODES + i] += 1.0f;
  __syncthreads();
  for (int i = tid; i < NODES; i += 256) {
    float s = 0.f;
    for (int j = 0; j < NODES; ++j) s += A[i * NODES + j];
    dinv[i] = (s > 0.f) ? rsqrtf(s) : 0.f;
  }
  __syncthreads();
  for (int idx = tid; idx < NODES * NODES; idx += 256) {
    int i = idx / NODES, j = idx % NODES;
    Ahat[idx] = dinv[i] * A[idx] * dinv[j];
  }
}

// ---------------- positional encoding table pe[T][D] ----------------
__global__ void build_pe(float* __restrict__ pe) {
  int idx = blockIdx.x * blockDim.x + threadIdx.x;
  if (idx >= TT * (DMODEL / 2)) return;
  int t = idx / (DMODEL / 2);
  int k = idx % (DMODEL / 2);
  float div = expf((float)(2 * k) * (-logf(10000.f) / (float)DMODEL));
  float ang = (float)t * div;
  pe[t * DMODEL + 2 * k]     = sinf(ang);
  pe[t * DMODEL + 2 * k + 1] = cosf(ang);
}

// ---------------- GCN: src2[tb, c*81+i] = (A_hat @ (xg@gw))[i,c] + gb[c] ----
__global__ void __launch_bounds__(128) gcn_kernel(const float* __restrict__ src,
                                                  const float* __restrict__ Ahat,
                                                  const float* __restrict__ gw,
                                                  const float* __restrict__ gb,
                                                  float* __restrict__ src2) {
  int tb = blockIdx.x;                       // 0..8191 (t*B+b)
  const float* s = src + (size_t)tb * IN_DIM;
  __shared__ float y0[NODES], y1[NODES];
  int tid = threadIdx.x;
  if (tid < NODES) {
    float x0 = s[tid];            // xg[node, c=0]
    float x1 = s[NODES + tid];    // xg[node, c=1]
    y0[tid] = x0 * gw[0] + x1 * gw[2];
    y1[tid] = x0 * gw[1] + x1 * gw[3];
  }
  __syncthreads();
  for (int idx = tid; idx < IN_DIM; idx += 128) {
    int c = idx / NODES, i = idx % NODES;
    const float* ar = Ahat + i * NODES;
    const float* yy = c ? y1 : y0;
    float sum = gb[c];
    for (int j = 0; j < NODES; ++j) sum += ar[j] * yy[j];
    src2[(size_t)tb * IN_DIM + c * NODES + i] = sum;
  }
}

// ---------------- generic tiled f32-WMMA GEMM ----------------
// C[M,N] = act( A[M,K] @ W[K,N] + bias[N] + add[(m/rowdiv)*N + n] )
// LDS layouts chosen so every WMMA fragment is ONE 8B-aligned ds_load_b64:
//   As[m][k]  : A pair (k, k+1) contiguous, row stride 34 floats (136B, 8B-aligned)
//   WsT[n][k] : W pair (k, k+1) contiguous after transpose-on-stage
#define GBM 64
#define GBN 64
#define GKT 32
#define KPAD (GKT + 2)
__global__ void __launch_bounds__(128) gemm_wmma_f32(
    const float* __restrict__ A, const float* __restrict__ W,
    const float* __restrict__ bias, const float* __restrict__ add, int rowdiv,
    int do_relu, float* __restrict__ C, int M, int N, int K) {
  __shared__ float As[GBM][KPAD];
  __shared__ float WsT[GBN][KPAD];
  int tid  = threadIdx.x;
  int m0   = blockIdx.y * GBM, n0 = blockIdx.x * GBN;
  int wave = tid >> 5, lane = tid & 31;
  int g    = lane >> 4, lm = lane & 15;
  int wrow = wave * 16;

  v8f zero8 = {0, 0, 0, 0, 0, 0, 0, 0};
  v8f acc[4] = {zero8, zero8, zero8, zero8};

  int nchunk = (K + GKT - 1) / GKT;
  for (int kc = 0; kc < nchunk; ++kc) {
    int k0 = kc * GKT;
    // stage A (64x32) and W^T (64x32), zero-padded at K/N tails
    for (int i = tid; i < GBM * GKT; i += 128) {
      int mm = i / GKT, kk = i % GKT;
      int gm = m0 + mm, gk = k0 + kk;
      As[mm][kk] = (gm < M && gk < K) ? A[(size_t)gm * K + gk] : 0.f;
    }
    for (int i = tid; i < GKT * GBN; i += 128) {
      int kk = i / GBN, nn = i % GBN;
      int gk = k0 + kk, gn = n0 + nn;
      WsT[nn][kk] = (gk < K && gn < N) ? W[(size_t)gk * N + gn] : 0.f;
    }
    // speculative prefetch of the next K-chunk (global_prefetch_b8)
    if (k0 + GKT < K) {
      __builtin_prefetch(&A[(size_t)(m0 + (tid >> 1)) * K + k0 + GKT], 0, 1);
      __builtin_prefetch(&W[(size_t)(k0 + GKT + (tid >> 2)) * N + n0], 0, 1);
    }
    __syncthreads();
    // preload all fragments of this chunk: batch of ds_load_b64, one wait,
    // then 32 back-to-back v_wmma_f32_16x16x4_f32
    v2f afr[8];
    v2f bfr[8][4];
#pragma unroll
    for (int s = 0; s < 8; ++s) {
      int kb = s * 4 + g * 2;          // lanes>=16 take K+2 (ISA A/B layout)
      afr[s] = *(const v2f*)&As[wrow + lm][kb];
#pragma unroll
      for (int c = 0; c < 4; ++c)
        bfr[s][c] = *(const v2f*)&WsT[c * 16 + lm][kb];
    }
#pragma unroll
    for (int s = 0; s < 8; ++s) {
#pragma unroll
      for (int c = 0; c < 4; ++c)
        acc[c] = wmma_f32(afr[s], bfr[s][c], acc[c]);
    }
    __syncthreads();
  }
  // epilogue: D layout row = j + 8*g, col = lane&15
  for (int c = 0; c < 4; ++c) {
    int gc = n0 + c * 16 + lm;
    if (gc >= N) continue;
    float bv = bias ? bias[gc] : 0.f;
#pragma unroll
    for (int j = 0; j < 8; ++j) {
      int gr = m0 + wrow + j + 8 * g;
      if (gr >= M) continue;
      float v = acc[c][j] + bv;
      if (add) v += add[(size_t)(gr / rowdiv) * N + gc];
      if (do_relu) v = v > 0.f ? v : 0.f;
      C[(size_t)gr * N + gc] = v;
    }
  }
}

// ---------------- layout permutes: [t*B+b, h*16+d] <-> [b,h,t,d] ----------
__global__ void to_heads(const float* __restrict__ in, float* __restrict__ out) {
  int idx = blockIdx.x * 256 + threadIdx.x;
  if (idx >= MROWS * DMODEL) return;
  int m = idx / DMODEL, n = idx % DMODEL;
  int t = m / BB, b = m % BB;
  int h = n / HD, d = n % HD;
  out[(((size_t)(b * NH + h)) * TT + t) * HD + d] = in[idx];
}
__global__ void from_heads(const float* __restrict__ in, float* __restrict__ out) {
  int idx = blockIdx.x * 256 + threadIdx.x;
  if (idx >= MROWS * DMODEL) return;
  int m = idx / DMODEL, n = idx % DMODEL;
  int t = m / BB, b = m % BB;
  int h = n / HD, d = n % HD;
  out[idx] = in[(((size_t)(b * NH + h)) * TT + t) * HD + d];
}

// ---------------- flash causal attention, one wave per 16-row q-tile -------
__global__ void __launch_bounds__(32) flash_attn(const float* __restrict__ Q,
                                                 const float* __restrict__ Km,
                                                 const float* __restrict__ V,
                                                 float* __restrict__ O) {
  int bh = blockIdx.y;     // b*H + h
  int qt = blockIdx.x;     // query tile (16 rows)
  const float* Qb = Q + (size_t)bh * TT * HD;
  const float* Kb = Km + (size_t)bh * TT * HD;
  const float* Vb = V + (size_t)bh * TT * HD;
  int lane = threadIdx.x;
  int g = lane >> 4, lm = lane & 15;

  // Q fragment (A layout), pre-scaled by 1/sqrt(HD)=0.25
  v2f qa[4];
#pragma unroll
  for (int s = 0; s < 4; ++s) {
    int kb = 4 * s + 2 * g;
    v2f q = *(const v2f*)(Qb + (size_t)(qt * 16 + lm) * HD + kb);
    qa[s] = q * 0.25f;
  }

  v8f zero8 = {0, 0, 0, 0, 0, 0, 0, 0};
  v8f o = zero8;
  float rmax[8], rsum[8];
#pragma unroll
  for (int j = 0; j < 8; ++j) { rmax[j] = -3.0e38f; rsum[j] = 0.f; }

  // P tile staged for C->A re-fragmenting; stride 18 floats (72B): pairs are
  // 8B-aligned ds_load_b64 and mod-64 banks stay conflict-free.
  __shared__ float Pt[16][18];

  for (int kt = 0; kt <= qt; ++kt) {
    // S = (Q*0.25) @ K^T : contraction over HD=16 in 4 WMMA steps
    v8f sa = zero8;
#pragma unroll
    for (int s = 0; s < 4; ++s) {
      int kb = 4 * s + 2 * g;
      v2f b = *(const v2f*)(Kb + (size_t)(kt * 16 + lm) * HD + kb);
      sa = wmma_f32(qa[s], b, sa);
    }
    bool diag = (kt == qt);
    // online softmax; row r of the D-tile = j + 8*g, its 16 cols live in the
    // lane half-group -> reduce with shfl_xor masks 1,2,4,8 (stays in-half)
#pragma unroll
    for (int j = 0; j < 8; ++j) {
      int row = j + 8 * g;
      float sv = sa[j];
      if (diag && lm > row) sv = -3.0e38f;   // causal mask: key > query
      float mx = sv;
      mx = fmaxf(mx, __shfl_xor(mx, 1, 32));
      mx = fmaxf(mx, __shfl_xor(mx, 2, 32));
      mx = fmaxf(mx, __shfl_xor(mx, 4, 32));
      mx = fmaxf(mx, __shfl_xor(mx, 8, 32));
      float nm = fmaxf(rmax[j], mx);
      float sc = __expf(rmax[j] - nm);
      float pv = __expf(sv - nm);
      float ps = pv;
      ps += __shfl_xor(ps, 1, 32);
      ps += __shfl_xor(ps, 2, 32);
      ps += __shfl_xor(ps, 4, 32);
      ps += __shfl_xor(ps, 8, 32);
      rsum[j] = rsum[j] * sc + ps;
      rmax[j] = nm;
      o[j] = o[j] * sc;
      Pt[row][lm] = pv;
    }
    __syncthreads();
    // O += P @ V  (P re-read from LDS in A layout; V fragment from global)
#pragma unroll
    for (int s = 0; s < 4; ++s) {
      int kb = 4 * s + 2 * g;
      v2f pa = *(const v2f*)&Pt[lm][kb];
      v2f vb;
      vb.x = Vb[(size_t)(kt * 16 + kb) * HD + lm];
      vb.y = Vb[(size_t)(kt * 16 + kb + 1) * HD + lm];
      o = wmma_f32(pa, vb, o);
    }
    __syncthreads();
  }
#pragma unroll
  for (int j = 0; j < 8; ++j) {
    int row = j + 8 * g;
    O[((size_t)bh * TT + qt * 16 + row) * HD + lm] = o[j] / rsum[j];
  }
}

// ---------------- LayerNorm over last dim (256) ----------------
__global__ void __launch_bounds__(256) layernorm_k(const float* __restrict__ in,
                                                   const float* __restrict__ w,
                                                   const float* __restrict__ b,
                                                   float* __restrict__ out) {
  int row = blockIdx.x;
  int tid = threadIdx.x;
  __shared__ float red[256];
  float v = in[(size_t)row * DMODEL + tid];
  red[tid] = v;
  __syncthreads();
  for (int s = 128; s > 0; s >>= 1) {
    if (tid < s) red[tid] += red[tid + s];
    __syncthreads();
  }
  float mean = red[0] / (float)DMODEL;
  __syncthreads();
  red[tid] = v * v;
  __syncthreads();
  for (int s = 128; s > 0; s >>= 1) {
    if (tid < s) red[tid] += red[tid + s];
    __syncthreads();
  }
  float var = red[0] / (float)DMODEL - mean * mean;
  float inv = rsqrtf(var + 1e-5f);
  out[(size_t)row * DMODEL + tid] = (v - mean) * inv * w[tid] + b[tid];
}

// ---------------- launch ----------------
extern "C" void kernel_launch(void* const* d_in, const int* in_sizes, int n_in,
                              void* d_out, int out_size, void* d_ws, size_t ws_size,
                              hipStream_t stream) {
  const float* src  = (const float*)d_in[0];
  const int*   ei   = (const int*)d_in[1];
  const float* gw   = (const float*)d_in[2];
  const float* gb   = (const float*)d_in[3];
  const float* fcw  = (const float*)d_in[4];
  const float* fcb  = (const float*)d_in[5];
  const float* wq   = (const float*)d_in[6],  *bq = (const float*)d_in[7];
  const float* wk   = (const float*)d_in[8],  *bk = (const float*)d_in[9];
  const float* wv   = (const float*)d_in[10], *bv = (const float*)d_in[11];
  const float* wo   = (const float*)d_in[12], *bo = (const float*)d_in[13];
  const float* ln1w = (const float*)d_in[14], *ln1b = (const float*)d_in[15];
  const float* f1w  = (const float*)d_in[16], *f1b  = (const float*)d_in[17];
  const float* f2w  = (const float*)d_in[18], *f2b  = (const float*)d_in[19];
  const float* ln2w = (const float*)d_in[20], *ln2b = (const float*)d_in[21];
  const float* dw   = (const float*)d_in[22], *db   = (const float*)d_in[23];
  int E = in_sizes[1] / 2;
  (void)n_in; (void)out_size; (void)ws_size;

  // bump allocator over d_ws (floats, 1KB-aligned chunks)
  float* base = (float*)d_ws;
  size_t off = 0;
  auto alloc = [&](size_t n) {
    float* p = base + off;
    off += (n + 255) & ~(size_t)255;
    return p;
  };
  float* ahat = alloc(NODES * NODES);
  float* pe   = alloc((size_t)TT * DMODEL);
  float* src2 = alloc((size_t)MROWS * IN_DIM);
  float* x    = alloc((size_t)MROWS * DMODEL);
  float* bufA = alloc((size_t)MROWS * DMODEL);
  float* bufB = alloc((size_t)MROWS * DMODEL);
  float* bufC = alloc((size_t)MROWS * DMODEL);
  float* qH   = alloc((size_t)MROWS * DMODEL);
  float* kH   = alloc((size_t)MROWS * DMODEL);
  float* vH   = alloc((size_t)MROWS * DMODEL);
  float* h1   = alloc((size_t)2048 * DFF);   // FFN hidden slab (2048 rows)

  dim3 blk128(128), blk256(256), blk32(32);
  dim3 gDD((DMODEL + 63) / 64, (MROWS + 63) / 64);
  int elemBlocks = (MROWS * DMODEL + 255) / 256;

  // 1) graph normalization + PE table + GCN feature build
  build_ahat<<<1, blk256, 0, stream>>>(ei, E, ahat);
  build_pe<<<(TT * (DMODEL / 2) + 255) / 256, blk256, 0, stream>>>(pe);
  gcn_kernel<<<MROWS, blk128, 0, stream>>>(src, ahat, gw, gb, src2);

  // 2) x = src2 @ fc_w + fc_b + pe[t]   (rowdiv=B maps m -> t)
  gemm_wmma_f32<<<gDD, blk128, 0, stream>>>(src2, fcw, fcb, pe, BB, 0, x,
                                            MROWS, DMODEL, IN_DIM);

  // 3) Q/K/V projections + head-major permute
  gemm_wmma_f32<<<gDD, blk128, 0, stream>>>(x, wq, bq, nullptr, 1, 0, bufA,
                                            MROWS, DMODEL, DMODEL);
  to_heads<<<elemBlocks, blk256, 0, stream>>>(bufA, qH);
  gemm_wmma_f32<<<gDD, blk128, 0, stream>>>(x, wk, bk, nullptr, 1, 0, bufB,
                                            MROWS, DMODEL, DMODEL);
  to_heads<<<elemBlocks, blk256, 0, stream>>>(bufB, kH);
  gemm_wmma_f32<<<gDD, blk128, 0, stream>>>(x, wv, bv, nullptr, 1, 0, bufC,
                                            MROWS, DMODEL, DMODEL);
  to_heads<<<elemBlocks, blk256, 0, stream>>>(bufC, vH);

  // 4) causal flash attention -> ctx (head layout in bufA slot)
  flash_attn<<<dim3(TT / 16, BB * NH), blk32, 0, stream>>>(qH, kH, vH, bufA);
  from_heads<<<elemBlocks, blk256, 0, stream>>>(bufA, bufB);

  // 5) ctx @ wo + bo + x, then LN1 -> x (reused)
  gemm_wmma_f32<<<gDD, blk128, 0, stream>>>(bufB, wo, bo, x, 1, 0, bufC,
                                            MROWS, DMODEL, DMODEL);
  layernorm_k<<<MROWS, blk256, 0, stream>>>(bufC, ln1w, ln1b, x);

  // 6) FFN in 4 row-slabs of 2048 (h1 scratch reuse): y2 -> bufC
  for (int sl = 0; sl < 4; ++sl) {
    size_t ro = (size_t)sl * 2048;
    dim3 g1(DFF / 64, 2048 / 64);
    gemm_wmma_f32<<<g1, blk128, 0, stream>>>(x + ro * DMODEL, f1w, f1b,
                                             nullptr, 1, 1, h1,
                                             2048, DFF, DMODEL);
    dim3 g2(DMODEL / 64, 2048 / 64);
    gemm_wmma_f32<<<g2, blk128, 0, stream>>>(h1, f2w, f2b, x + ro * DMODEL, 1,
                                             0, bufC + ro * DMODEL,
                                             2048, DMODEL, DFF);
  }
  layernorm_k<<<MROWS, blk256, 0, stream>>>(bufC, ln2w, ln2b, bufB);

  // 7) decoder: out[M,162] = x3 @ dec_w + dec_b
  dim3 gDec((IN_DIM + 63) / 64, (MROWS + 63) / 64);
  gemm_wmma_f32<<<gDec, blk128, 0, stream>>>(bufB, dw, db, nullptr, 1, 0,
                                             (float*)d_out, MROWS, IN_DIM,
                                             DMODEL);
}